// LayeredSuperModel_47390669144596
// MI455X (gfx1250) — compile-verified
//
#include <hip/hip_runtime.h>

#define NQ 12
#define DIM 4096          // 2^12 amplitudes per state
#define NCLS 10
#define NFEAT 36

typedef float v2f __attribute__((ext_vector_type(2)));
typedef float v8f __attribute__((ext_vector_type(8)));

// ---------------------------------------------------------------------------
// Prep: build the six fused 16x16 complex group unitaries
//   G[layer][grp] = kron(U_{4g}, U_{4g+1}, U_{4g+2}, U_{4g+3}),
//   U_w = RZ(az_w) * RY(ay_w) * RX(ax_w)   (per-layer fused 1-qubit gates)
// Workspace layout (floats): per (layer*3+grp): [256 real][256 imag], row-major 16x16.
// ---------------------------------------------------------------------------
__global__ __launch_bounds__(256) void qprep(const float* __restrict__ angles,
                                             float* __restrict__ G) {
  __shared__ float ur[4][2][2], ui[4][2][2];
  const int layer = blockIdx.x / 3;
  const int grp   = blockIdx.x % 3;
  const int tid   = threadIdx.x;

  if (tid < 4) {
    const int w = grp * 4 + tid;
    const float ax = angles[layer * 48 + w];
    const float ay = angles[layer * 48 + 12 + w];
    const float az = angles[layer * 48 + 24 + w];
    float cx, sx, cy, sy, cz, sz;
    sincosf(0.5f * ax, &sx, &cx);
    sincosf(0.5f * ay, &sy, &cy);
    sincosf(0.5f * az, &sz, &cz);
    // M1 = RY * RX
    const float m00r = cy * cx, m00i =  sy * sx;
    const float m01r = -sy * cx, m01i = -cy * sx;
    const float m10r =  sy * cx, m10i = -cy * sx;
    const float m11r =  cy * cx, m11i = -sy * sx;
    // U = RZ * M1 : row0 *= (cz - i sz), row1 *= (cz + i sz)
    ur[tid][0][0] = cz * m00r + sz * m00i;  ui[tid][0][0] = cz * m00i - sz * m00r;
    ur[tid][0][1] = cz * m01r + sz * m01i;  ui[tid][0][1] = cz * m01i - sz * m01r;
    ur[tid][1][0] = cz * m10r - sz * m10i;  ui[tid][1][0] = cz * m10i + sz * m10r;
    ur[tid][1][1] = cz * m11r - sz * m11i;  ui[tid][1][1] = cz * m11i + sz * m11r;
  }
  __syncthreads();

  // kron: wire (4g+q) is bit (3-q) of the 4-bit group index
  const int r = tid >> 4, c = tid & 15;
  float pr = 1.f, pi = 0.f;
  for (int q = 0; q < 4; ++q) {
    const int rb = (r >> (3 - q)) & 1, cb = (c >> (3 - q)) & 1;
    const float wr = ur[q][rb][cb], wi = ui[q][rb][cb];
    const float nr = pr * wr - pi * wi;
    const float ni = pr * wi + pi * wr;
    pr = nr; pi = ni;
  }
  float* base = G + (size_t)blockIdx.x * 512;
  base[tid]       = pr;
  base[256 + tid] = pi;
}

// ---------------------------------------------------------------------------
// Main: one workgroup (8 wave32) per batch state; state lives in LDS (re/im split).
// ---------------------------------------------------------------------------
__global__ __launch_bounds__(256) void qsim(const float* __restrict__ sv,
                                            const float* __restrict__ angles,
                                            const float* __restrict__ W,
                                            const float* __restrict__ bvec,
                                            const float* __restrict__ G,
                                            float* __restrict__ out) {
  __shared__ float re[DIM];
  __shared__ float im[DIM];
  __shared__ float part[NFEAT * 8];
  __shared__ float feat[NFEAT];

  const int tid  = threadIdx.x;
  const int b    = blockIdx.x;
  const int lane = tid & 31;
  const int wave = tid >> 5;
  const int n    = lane & 15;   // column / M-low index within the 16-lane half
  const int hi   = lane >> 4;   // half-wave select (ISA WMMA layout)

  // ---- load state: real part from input, imag = 0 (vectorized, coalesced) ----
  {
    const float4* sv4 = (const float4*)(sv + (size_t)b * DIM);
    for (int k = 0; k < 4; ++k) {
      const int i4 = k * 256 + tid;
      const float4 v = sv4[i4];
      re[i4 * 4 + 0] = v.x; re[i4 * 4 + 1] = v.y;
      re[i4 * 4 + 2] = v.z; re[i4 * 4 + 3] = v.w;
      im[i4 * 4 + 0] = 0.f; im[i4 * 4 + 1] = 0.f;
      im[i4 * 4 + 2] = 0.f; im[i4 * 4 + 3] = 0.f;
    }
  }
  __syncthreads();

  for (int layer = 0; layer < 2; ++layer) {
    // ---- fused 1-qubit gates: 3 x (16x16 complex matmul) via WMMA f32 16x16x4 ----
    for (int grp = 0; grp < 3; ++grp) {
      const float* Gr = G + (size_t)(layer * 3 + grp) * 512;
      const float* Gi = Gr + 256;

      // A-matrix chunks (16x4 each), ISA layout: v0 -> K=0 (lanes0-15)/K=2 (16-31), v1 -> K=1/K=3
      v2f Ar[4], Ai[4], An[4];
      for (int kk = 0; kk < 4; ++kk) {
        const int col = 4 * kk + 2 * hi;
        Ar[kk].x = Gr[n * 16 + col];  Ar[kk].y = Gr[n * 16 + col + 1];
        Ai[kk].x = Gi[n * 16 + col];  Ai[kk].y = Gi[n * 16 + col + 1];
        An[kk].x = -Ai[kk].x;         An[kk].y = -Ai[kk].y;   // f32 WMMA has no A-neg modifier
      }

      // LDS strides: amplitude offset = k*sk + o*so + j*sj   (k = acted 4-wire index)
      int sk, so, sj;
      if (grp == 0)      { sk = 256; so = 1;  sj = 16;  }  // wires 0-3  (top bits)
      else if (grp == 1) { sk = 16;  so = 1;  sj = 256; }  // wires 4-7  (mid bits)
      else               { sk = 1;   so = 16; sj = 256; }  // wires 8-11 (low bits)

      for (int t = 0; t < 2; ++t) {
        const int j = wave + 8 * t;              // this wave's independent 16x16 tile
        v2f Br[4], Bi[4];
        for (int kk = 0; kk < 4; ++kk) {
          const int off0 = j * sj + (4 * kk + 2 * hi) * sk + n * so;
          Br[kk].x = re[off0]; Br[kk].y = re[off0 + sk];
          Bi[kk].x = im[off0]; Bi[kk].y = im[off0 + sk];
        }
        v8f dr = {}; v8f di = {};
        for (int kk = 0; kk < 4; ++kk) {
          // Dr = Ar*Br - Ai*Bi ; Di = Ar*Bi + Ai*Br
          dr = __builtin_amdgcn_wmma_f32_16x16x4_f32(false, Ar[kk], false, Br[kk],
                                                     (short)0, dr, false, false);
          dr = __builtin_amdgcn_wmma_f32_16x16x4_f32(false, An[kk], false, Bi[kk],
                                                     (short)0, dr, false, false);
          di = __builtin_amdgcn_wmma_f32_16x16x4_f32(false, Ar[kk], false, Bi[kk],
                                                     (short)0, di, false, false);
          di = __builtin_amdgcn_wmma_f32_16x16x4_f32(false, Ai[kk], false, Br[kk],
                                                     (short)0, di, false, false);
        }
        // D layout: vgpr v holds D[v + 8*hi][n]
        for (int v = 0; v < 8; ++v) {
          const int off = j * sj + (v + 8 * hi) * sk + n * so;
          re[off] = dr[v];
          im[off] = di[v];
        }
      }
      __syncthreads();
    }

    // ---- CRX ring (order-dependent, sequential with barriers) ----
    for (int s = 0; s < 12; ++s) {
      const int i   = (layer == 0) ? s : (11 - s);
      const int c   = i;
      const int tg  = (i + 1) % 12;
      const float th = angles[(layer == 0) ? (36 + i) : (84 + (11 - i))];
      float cs, sn;
      sincosf(0.5f * th, &sn, &cs);
      const int scS = 1 << (11 - c);
      const int stS = 1 << (11 - tg);
      const int loS = (scS < stS) ? scS : stS;
      const int hiS = (scS < stS) ? stS : scS;
      for (int p = tid; p < 1024; p += 256) {
        int m     = ((p & ~(loS - 1)) << 1) | (p & (loS - 1));  // insert 0 at low bit
        int base2 = ((m & ~(hiS - 1)) << 1) | (m & (hiS - 1));  // insert 0 at high bit
        const int i0 = base2 + scS;      // control = 1, target = 0
        const int i1 = i0 + stS;         // control = 1, target = 1
        const float a0r = re[i0], a0i = im[i0];
        const float a1r = re[i1], a1i = im[i1];
        re[i0] = cs * a0r + sn * a1i;
        im[i0] = cs * a0i - sn * a1r;
        re[i1] = cs * a1r + sn * a0i;
        im[i1] = cs * a1i - sn * a0r;
      }
      __syncthreads();
    }
  }

  // ---- expectation values <X_q>, <Y_q>, <Z_q> (deterministic reduction) ----
  for (int q = 0; q < 12; ++q) {
    const int stq = 1 << (11 - q);
    float sx = 0.f, sy = 0.f, sz = 0.f;
    for (int p = tid; p < 2048; p += 256) {
      const int i0 = ((p & ~(stq - 1)) << 1) | (p & (stq - 1));
      const int i1 = i0 + stq;
      const float r0 = re[i0], q0 = im[i0];
      const float r1 = re[i1], q1 = im[i1];
      sz += r0 * r0 + q0 * q0 - r1 * r1 - q1 * q1;
      sx += 2.f * (r0 * r1 + q0 * q1);   // 2*Re(conj(p0)*p1)
      sy += 2.f * (r0 * q1 - q0 * r1);   // 2*Im(conj(p0)*p1)
    }
    for (int off = 16; off > 0; off >>= 1) {
      sx += __shfl_down(sx, off, 32);
      sy += __shfl_down(sy, off, 32);
      sz += __shfl_down(sz, off, 32);
    }
    if (lane == 0) {
      part[q * 8 + wave]        = sx;
      part[(12 + q) * 8 + wave] = sy;
      part[(24 + q) * 8 + wave] = sz;
    }
  }
  __syncthreads();
  if (tid < NFEAT) {
    float a = 0.f;
    for (int w = 0; w < 8; ++w) a += part[tid * 8 + w];
    feat[tid] = a;
  }
  __syncthreads();

  // ---- linear head: out[b, cls] = feat . W[cls] + b[cls] ----
  if (tid < NCLS) {
    float acc = bvec[tid];
    for (int f = 0; f < NFEAT; ++f) acc += feat[f] * W[tid * NFEAT + f];
    out[(size_t)b * NCLS + tid] = acc;
  }
}

// ---------------------------------------------------------------------------
extern "C" void kernel_launch(void* const* d_in, const int* in_sizes, int n_in,
                              void* d_out, int out_size, void* d_ws, size_t ws_size,
                              hipStream_t stream) {
  const float* sv     = (const float*)d_in[0];   // [2048, 4096] f32
  const float* angles = (const float*)d_in[1];   // [96] f32
  const float* W      = (const float*)d_in[2];   // [10, 36] f32
  const float* bvec   = (const float*)d_in[3];   // [10] f32
  float* out = (float*)d_out;                    // [2048, 10] f32
  float* G   = (float*)d_ws;                     // 6 * 512 floats = 12 KB

  const int batch = in_sizes[0] / DIM;

  qprep<<<6, 256, 0, stream>>>(angles, G);
  qsim<<<batch, 256, 0, stream>>>(sv, angles, W, bvec, G, out);
}